// InductiveNodeEncoder_39247411151457
// MI455X (gfx1250) — compile-verified
//
#include <hip/hip_runtime.h>
#include <hip/hip_bf16.h>

typedef __attribute__((ext_vector_type(2))) float v2f;
typedef __attribute__((ext_vector_type(8))) float v8f;

#define APAD 260   // 16x256 A' tile pitch (pad 4 -> conflict-free WMMA frag reads)
#define HPAD 132   // 16x128 H tile pitch

// ---------------------------------------------------------------- zero fill
__global__ void zero_f32(float* __restrict__ p, size_t count) {
    size_t i = (size_t)blockIdx.x * blockDim.x + threadIdx.x;
    if (i < count) p[i] = 0.0f;
}

// ------------------------------------------------- edge gather/scatter-add
// one wave (32 lanes) per edge; lane handles 4 consecutive features (float4)
__global__ __launch_bounds__(256) void scatter_add(
    const float* __restrict__ feat, const int* __restrict__ ei,
    float* __restrict__ agg, float* __restrict__ deg, int E) {
    int wave = (int)((blockIdx.x * (size_t)blockDim.x + threadIdx.x) >> 5);
    int lane = threadIdx.x & 31;
    if (wave >= E) return;
    int src = ei[wave];          // edge_index[0][e]
    int dst = ei[(size_t)E + wave]; // edge_index[1][e]
    const float4 v = *(const float4*)(feat + (size_t)src * 128 + lane * 4);
    float* dp = agg + (size_t)dst * 128 + lane * 4;
    atomicAdd(dp + 0, v.x);
    atomicAdd(dp + 1, v.y);
    atomicAdd(dp + 2, v.z);
    atomicAdd(dp + 3, v.w);
    if (deg != nullptr && lane == 0) atomicAdd(deg + dst, 1.0f);
}

// ---------------------------------------------------------- fused SAGE GEMM
// out[i,:] = l2norm( mean[i,:] @ Wl + xself[i,:] @ Wr + b ), mean = agg/max(deg,1)
// one workgroup = 16-row tile; 8 waves x one 16-col WMMA tile each (128 cols).
// A' = [mean | xself] (16x256) staged in LDS; K=256 in steps of 4 f32 WMMAs.
// do_stats: also accumulate per-column sum / sum-of-squares (for BatchNorm).
__global__ __launch_bounds__(256) void sage_gemm_norm(
    const float* __restrict__ xself, const float* __restrict__ agg,
    const float* __restrict__ deg,  const float* __restrict__ Wl,
    const float* __restrict__ Wr,   const float* __restrict__ bias,
    float* __restrict__ out, float* __restrict__ colsum, float* __restrict__ colsq,
    int n, int do_stats) {
    __shared__ float As[16 * APAD];
    __shared__ float Hs[16 * HPAD];
    __shared__ float rdeg[16];
    __shared__ float rowsq[16];
    __shared__ float cstat[256];

    const int t  = threadIdx.x;
    const int i0 = blockIdx.x * 16;

    if (t < 16) {
        int row = i0 + t;
        float d = (row < n) ? deg[row] : 1.0f;
        rdeg[t]  = 1.0f / fmaxf(d, 1.0f);
        rowsq[t] = 0.0f;
    }
    cstat[t] = 0.0f;
    __syncthreads();

    // stage A' tile: cols 0..127 = agg*rdeg (mean), 128..255 = self features
    for (int idx = t; idx < 16 * 256; idx += 256) {
        int r = idx >> 8;
        int k = idx & 255;
        int row = i0 + r;
        float v = 0.0f;
        if (row < n) {
            v = (k < 128) ? agg[(size_t)row * 128 + k] * rdeg[r]
                          : xself[(size_t)row * 128 + (k - 128)];
        }
        As[r * APAD + k] = v;
    }
    __syncthreads();

    // ---- WMMA: D = A'(16x256) x W'(256x128), per-wave 16x16 tile
    const int wave = t >> 5;
    const int lane = t & 31;
    const int half = lane >> 4;   // 0: K, K+1 ; 1: K+2, K+3
    const int m    = lane & 15;
    const int col  = wave * 16 + m;

    v8f c = {0.f, 0.f, 0.f, 0.f, 0.f, 0.f, 0.f, 0.f};
    for (int k = 0; k < 256; k += 4) {
        const int kk = k + 2 * half;                  // even
        v2f a = *(const v2f*)&As[m * APAD + kk];      // A frag: rows m, K=kk..kk+1
        const float* Wsel = (kk < 128) ? Wl : Wr;     // W' = [Wl ; Wr]
        const int kw = kk & 127;
        v2f b;
        b.x = Wsel[(size_t)kw * 128 + col];           // B frag: K=kk..kk+1, N=col
        b.y = Wsel[(size_t)(kw + 1) * 128 + col];
        c = __builtin_amdgcn_wmma_f32_16x16x4_f32(
                /*neg_a=*/false, a, /*neg_b=*/false, b,
                /*c_mod=*/(short)0, c, /*reuse_a=*/false, /*reuse_b=*/false);
    }

    // spill 16x16 tile (+bias) to LDS: VGPR v -> row v + 8*half, col = N
    const float bj = bias[col];
#pragma unroll
    for (int v = 0; v < 8; ++v) {
        Hs[(v + 8 * half) * HPAD + col] = c[v] + bj;
    }
    __syncthreads();

    // row sum-of-squares (16 threads/row, 8 elems each, LDS atomic reduce)
    {
        int r = t >> 4, g = t & 15;
        float s = 0.0f;
#pragma unroll
        for (int q = 0; q < 8; ++q) {
            float h = Hs[r * HPAD + g * 8 + q];
            s += h * h;
        }
        atomicAdd(&rowsq[r], s);
    }
    __syncthreads();

    // l2-normalize, store, accumulate column stats
    {
        int r = t >> 4, g = t & 15;
        int row = i0 + r;
        if (row < n) {
            float rn = 1.0f / fmaxf(sqrtf(rowsq[r]), 1e-12f);
#pragma unroll
            for (int q = 0; q < 8; ++q) {
                int cc = g * 8 + q;
                float v = Hs[r * HPAD + cc] * rn;
                out[(size_t)row * 128 + cc] = v;
                if (do_stats) {
                    atomicAdd(&cstat[cc], v);
                    atomicAdd(&cstat[128 + cc], v * v);
                }
            }
        }
    }
    if (do_stats) {
        __syncthreads();
        if (t < 128)      atomicAdd(&colsum[t], cstat[t]);
        else              atomicAdd(&colsq[t - 128], cstat[t]);
    }
}

// --------------------------------------------- BatchNorm stats -> scale/shift
__global__ void bn_prepare(const float* __restrict__ colsum, const float* __restrict__ colsq,
                           const float* __restrict__ gamma,  const float* __restrict__ beta,
                           float* __restrict__ scale, float* __restrict__ shift, float invN) {
    int c = threadIdx.x;  // 128 threads
    float mu  = colsum[c] * invN;
    float var = colsq[c] * invN - mu * mu;
    float s   = gamma[c] * rsqrtf(var + 1e-5f);
    scale[c]  = s;
    shift[c]  = beta[c] - mu * s;
}

// ------------------------------------------------------ BN apply + ReLU (in place)
__global__ void bn_relu(float* __restrict__ H, const float* __restrict__ scale,
                        const float* __restrict__ shift, size_t count) {
    size_t idx = (size_t)blockIdx.x * blockDim.x + threadIdx.x;
    if (idx < count) {
        int c = (int)(idx & 127);
        float v = H[idx] * scale[c] + shift[c];
        H[idx] = fmaxf(v, 0.0f);
    }
}

extern "C" void kernel_launch(void* const* d_in, const int* in_sizes, int n_in,
                              void* d_out, int out_size, void* d_ws, size_t ws_size,
                              hipStream_t stream) {
    const float* x      = (const float*)d_in[0];
    const int*   ei     = (const int*)  d_in[1];
    const float* Wl0    = (const float*)d_in[2];
    const float* Wr0    = (const float*)d_in[3];
    const float* b0     = (const float*)d_in[4];
    const float* gamma0 = (const float*)d_in[5];
    const float* beta0  = (const float*)d_in[6];
    const float* Wl1    = (const float*)d_in[7];
    const float* Wr1    = (const float*)d_in[8];
    const float* b1     = (const float*)d_in[9];
    float* out = (float*)d_out;

    const int n = in_sizes[0] / 128;   // num nodes
    const int E = in_sizes[1] / 2;     // num edges

    // workspace layout (floats): agg | deg | colsum | colsq | bnscale | bnshift | H0
    float* ws      = (float*)d_ws;
    float* agg     = ws;
    float* deg     = agg + (size_t)n * 128;
    float* colsum  = deg + n;
    float* colsq   = colsum + 128;
    float* bnscale = colsq + 128;
    float* bnshift = bnscale + 128;
    float* H0      = bnshift + 128;

    const size_t nf      = (size_t)n * 128;
    const size_t zcount  = nf + (size_t)n + 256;          // agg+deg+stats contiguous
    const int    sblocks = (int)(((size_t)E * 32 + 255) / 256);
    const int    tiles   = (n + 15) / 16;

    // ---- layer 0
    zero_f32<<<(int)((zcount + 255) / 256), 256, 0, stream>>>(agg, zcount);
    scatter_add<<<sblocks, 256, 0, stream>>>(x, ei, agg, deg, E);
    sage_gemm_norm<<<tiles, 256, 0, stream>>>(x, agg, deg, Wl0, Wr0, b0,
                                              H0, colsum, colsq, n, 1);
    bn_prepare<<<1, 128, 0, stream>>>(colsum, colsq, gamma0, beta0,
                                      bnscale, bnshift, 1.0f / (float)n);
    bn_relu<<<(int)((nf + 255) / 256), 256, 0, stream>>>(H0, bnscale, bnshift, nf);

    // ---- layer 1 (deg reused)
    zero_f32<<<(int)((nf + 255) / 256), 256, 0, stream>>>(agg, nf);
    scatter_add<<<sblocks, 256, 0, stream>>>(H0, ei, agg, nullptr, E);
    sage_gemm_norm<<<tiles, 256, 0, stream>>>(H0, agg, deg, Wl1, Wr1, b1,
                                              out, nullptr, nullptr, n, 0);
}